// RotaryPositionalEmbedding_12326556139778
// MI455X (gfx1250) — compile-verified
//
#include <hip/hip_runtime.h>
#include <stdint.h>

// --- ext-vector types matching the amdgpu-toolchain (clang-23) TDM builtin:
//     __builtin_amdgcn_tensor_load_to_lds(uint32x4 g0, int32x8 g1,
//                                         int32x4, int32x4, int32x8, i32 cpol)
typedef unsigned int u32x4 __attribute__((ext_vector_type(4)));
typedef int          i32x8 __attribute__((ext_vector_type(8)));
typedef int          i32x4 __attribute__((ext_vector_type(4)));
typedef float        f32x2 __attribute__((ext_vector_type(2)));

#define DK              64                 // head dim (reference D_K)
#define ROWS_PER_BLOCK  8                  // one wave32 per row
#define BLOCK_THREADS   (ROWS_PER_BLOCK * 32)

// RoPE via block-diagonal structure:
//   out[2k]   = c*x[2k] - s*x[2k+1]
//   out[2k+1] = s*x[2k] + c*x[2k+1]
// (c, -s) are ADJACENT floats at R[p, 2k, 2k] / R[p, 2k, 2k+1]
// -> one aligned 8-byte gather per lane (pair k <-> lane k, wave32 == 32 pairs).
//
// Cache policy: rope buffer (128 MB, ~4x reuse) is kept L2-resident (default RT);
// x is streamed in via TDM with TH=NT; out is stored non-temporally. This keeps
// the 192 MB L2 dedicated to the gathered coefficients.
__global__ __launch_bounds__(BLOCK_THREADS)
void rope_tdm_kernel(const float* __restrict__ x,
                     const int*   __restrict__ tp,
                     const float* __restrict__ R,
                     float*       __restrict__ out,
                     int n_rows)
{
    __shared__ float xs[ROWS_PER_BLOCK * DK];   // 2 KB x staging tile (TDM target)

    const int tid  = threadIdx.x;
    const int wave = tid >> 5;
    const int lane = tid & 31;
    const int row0 = blockIdx.x * ROWS_PER_BLOCK;
    const int row  = row0 + wave;

    // ---------- rotation-coefficient gather (overlaps the TDM transfer) ----------
    float c = 0.0f, ns = 0.0f;  // ns = -sin
    if (row < n_rows) {
        const int p = tp[row];  // wave-uniform position
        // element offset inside matrix p: (2*lane)*64 + 2*lane = 130*lane
        // byte offset 520*lane -> 8-byte aligned; loads (cos, -sin) in one b64.
        const float* rrow = R + (size_t)p * (DK * DK) + (size_t)lane * (2 * DK + 2);
        const float2 cs = *(const float2*)rrow;
        c  = cs.x;
        ns = cs.y;
    }

    // ---------- TDM: async DMA of this block's contiguous x tile into LDS ----------
    // Issued once by wave 0 (TDM ignores EXEC; branch is wave-uniform).
    if (wave == 0) {
        int tile_rows  = n_rows - row0;
        if (tile_rows > ROWS_PER_BLOCK) tile_rows = ROWS_PER_BLOCK;
        const unsigned tile_elems = (unsigned)tile_rows * DK;     // <= 512 f32

        const uint64_t ga = (uint64_t)(uintptr_t)(x + (size_t)row0 * DK);
        // Flat shared-aperture address: low 32 bits ARE the LDS byte offset
        // (ISA 10.2 aperture mapping).
        const uint32_t la = (uint32_t)(uintptr_t)&xs[0];

        // ----- D# group 0 (128b): count=1 valid user desc, lds_addr, global_addr, type=2
        u32x4 g0;
        g0.x = 1u;                                               // count=1
        g0.y = la;                                               // lds_addr [63:32]
        g0.z = (uint32_t)ga;                                     // global_addr[31:0]
        g0.w = (uint32_t)((ga >> 32) & 0x01FFFFFFu)              // global_addr[56:32]
             | 0x80000000u;                                      // type=2 ("image")

        // ----- D# group 1 (256b): data_size=4B, 1-D tile of tile_elems elements
        i32x8 g1;
        g1[0] = 0x00020000;                       // data_size=2 (4B); no multicast/pad/iter
        g1[1] = (int)((tile_elems & 0xFFFFu) << 16);            // tensor_dim0[15:0]
        g1[2] = (int)((tile_elems >> 16) | (1u << 16));         // tensor_dim0[31:16] | tensor_dim1=1
        g1[3] = (int)(tile_elems << 16);                        // tile_dim0 (<=512)
        g1[4] = 0;                                              // tile_dim1=0, tile_dim2=0 (1-D)
        g1[5] = (int)tile_elems;                                // tensor_dim0_stride
        g1[6] = 0;
        g1[7] = 0;

        const i32x4 z4 = {0, 0, 0, 0};                          // groups 2/3 unused (<=2D)
        const i32x8 z8 = {0, 0, 0, 0, 0, 0, 0, 0};              // extra group (clang-23 form)
        // cpol = 1 -> TH[2:0]=NT: x is streamed once, keep it out of L2's
        // rope-coefficient working set.
        __builtin_amdgcn_tensor_load_to_lds(g0, g1, z4, z4, z8, 1);
        __builtin_amdgcn_s_wait_tensorcnt(0);                    // s_wait_tensorcnt 0
    }
    __syncthreads();                                             // release other 7 waves

    if (row >= n_rows) return;

    // ---------- compute + non-temporal store ----------
    // ds_load_b64: lane k reads banks {2k, 2k+1} -> 64 banks covered, conflict-free.
    const float2 xv = *(const float2*)&xs[wave * DK + 2 * lane];
    f32x2 o;
    o.x = fmaf(c, xv.x,  ns * xv.y);    //  c*x0 - s*x1
    o.y = fmaf(c, xv.y, -ns * xv.x);    //  s*x0 + c*x1
    f32x2* optr = (f32x2*)(out + (size_t)row * DK + 2 * lane);
    __builtin_nontemporal_store(o, optr);   // global_store_b64 ... th:TH_NT
}

extern "C" void kernel_launch(void* const* d_in, const int* in_sizes, int n_in,
                              void* d_out, int out_size, void* d_ws, size_t ws_size,
                              hipStream_t stream)
{
    const float* x   = (const float*)d_in[0];   // (4, 8192, 64) f32
    const int*   tp  = (const int*)  d_in[1];   // (4, 8192)     i32
    const float* R   = (const float*)d_in[2];   // (8192, 64, 64) f32
    float*       out = (float*)d_out;

    const int n_rows = in_sizes[1];             // B*S = 32768 rows of 64 floats
    const int blocks = (n_rows + ROWS_PER_BLOCK - 1) / ROWS_PER_BLOCK;

    rope_tdm_kernel<<<blocks, BLOCK_THREADS, 0, stream>>>(x, tp, R, out, n_rows);
}